// RiemannResonanceEngine_76647986365160
// MI455X (gfx1250) — compile-verified
//
#include <hip/hip_runtime.h>

// ---------------------------------------------------------------------------
// Persistent LDS-resident leapfrog wave solver for gfx1250 (MI455X).
// Laplacian on the matrix pipe:  lap = T*U + U*T + corner terms,
// 12x V_WMMA_F32_16X16X4_F32 per 16x16 tile (exact f32 arithmetic).
// 8 workgroups (4 row-bands x 2 images), halo rows exchanged via L2 with a
// monotonic release/acquire flag protocol (deterministic, reset via memset).
// Melt and probe phases are separately specialized loops (no per-step branch).
// ---------------------------------------------------------------------------

typedef float v2f __attribute__((ext_vector_type(2)));
typedef float v8f __attribute__((ext_vector_type(8)));

#define MELT_STEPS  300
#define PROBE_STEPS 400
#define TOTAL_STEPS (MELT_STEPS + PROBE_STEPS)
#define H     256
#define IMG   (H*H)
#define WGS   8            // 4 row-bands x 2 images
#define BAND  64           // interior rows per workgroup
#define THREADS 512        // 16 waves (wave32), 4 tiles per wave
#define PITCH 264          // 4 ghost + 256 + 4 ghost cols (b64/b128 aligned)
#define ROWS  72           // 4 ghost + 64 + 4 ghost rows
#define BUFN  (ROWS*PITCH)

// ws layout (32-bit words):
//   [0..7]                      per-WG step flags (memset to 0 each call)
//   [8 .. 8+WGS*400)            per-WG energy partials
//   [WS_SLOTS .. +WGS*2*2*256)  halo slots [wg][side][parity][256]
// total = 11400 words = 45.6 KB of d_ws
#define WS_ENERGY 8
#define WS_SLOTS  (8 + WGS*PROBE_STEPS)

__device__ __forceinline__ float tmat(int i, int j) {
  int d = i - j; d = d < 0 ? -d : d;            // tridiagonal 2nd difference
  return d == 0 ? -2.0f : (d == 1 ? 1.0f : 0.0f);
}

__device__ __forceinline__ v8f wmma4(v2f a, v2f b, v8f c) {
  // V_WMMA_F32_16X16X4_F32 : D = A(16x4) x B(4x16) + C(16x16), all f32
  return __builtin_amdgcn_wmma_f32_16x16x4_f32(false, a, false, b,
                                               (short)0, c, false, false);
}

// One leapfrog step: WMMA Laplacian, pointwise update, halo exchange.
template <bool PROBE>
__device__ __forceinline__ void do_step(
    int g, int wg, int tid, int lane, int wv, int lo, int hi,
    int hasUp, int hasDn, int r0, int Jb,
    float* rd, float* wr, float* scr,
    const v2f (&Ta)[4], v2f EP, v2f EQ, v8f (&ml)[4],
    unsigned int* flags, float* energy, float* slots)
{
  float e = 0.0f;
#pragma unroll
  for (int t = 0; t < 4; ++t) {
    const int c0 = 4 + (Jb + t)*16;
    v8f acc = (v8f){0,0,0,0,0,0,0,0};

    // T * U(I,J) : 4 WMMAs (B-operand = row chunks of the tile)
#pragma unroll
    for (int c = 0; c < 4; ++c) {
      v2f ub;
      ub.x = rd[(r0 + 4*c + 2*hi    )*PITCH + c0 + lo];
      ub.y = rd[(r0 + 4*c + 2*hi + 1)*PITCH + c0 + lo];
      acc = wmma4(Ta[c], ub, acc);
    }
    // U(I,J) * T : 4 WMMAs (A-operand = col chunks, b64 loads)
#pragma unroll
    for (int c = 0; c < 4; ++c) {
      v2f ua = *(const v2f*)&rd[(r0 + lo)*PITCH + c0 + 4*c + 2*hi];
      acc = wmma4(ua, Ta[c], acc);
    }
    { // row above: P * U(I-1,J)  (only its last row contributes)
      v2f nb;
      nb.x = rd[(r0 - 4 + 2*hi)*PITCH + c0 + lo];
      nb.y = rd[(r0 - 3 + 2*hi)*PITCH + c0 + lo];
      acc = wmma4(EP, nb, acc);
    }
    { // row below: Q * U(I+1,J)
      v2f nb;
      nb.x = rd[(r0 + 16 + 2*hi)*PITCH + c0 + lo];
      nb.y = rd[(r0 + 17 + 2*hi)*PITCH + c0 + lo];
      acc = wmma4(EQ, nb, acc);
    }
    { // col left: U(I,J-1) * Q
      v2f la = *(const v2f*)&rd[(r0 + lo)*PITCH + c0 - 4 + 2*hi];
      acc = wmma4(la, EP, acc);
    }
    { // col right: U(I,J+1) * P
      v2f ra = *(const v2f*)&rd[(r0 + lo)*PITCH + c0 + 16 + 2*hi];
      acc = wmma4(ra, EQ, acc);
    }

    // pointwise leapfrog update in the WMMA C/D lane layout
#pragma unroll
    for (int r = 0; r < 8; ++r) {
      const int off = (r0 + r + 8*hi)*PITCH + c0 + lo;
      float u   = rd[off];
      float up  = wr[off];                      // u(g-2) center
      float lap = acc[r];
      float un;
      if (PROBE) {
        un = 0.995f * (2.0f*u - up + ml[t][r] * lap);   // ml holds c^2
        e += un * un;
      } else {
        un = 0.98f * (2.0f*u - up + 0.24f * lap);
        float al = fabsf(lap);
        if (al > ml[t][r]) ml[t][r] = al;               // running max|lap|
      }
      wr[off] = un;                                     // becomes u(g)
    }
  }

  if (PROBE) {                                  // deterministic wave reduce
    for (int off = 16; off; off >>= 1) e += __shfl_xor(e, off, 32);
    if (lane == 0) scr[wv] = e;
  }
  __syncthreads();                              // LDS writes + scr visible

  const int par = g & 1;
  if (tid < 256) {                              // export first interior row
    slots[((wg*2 + 0)*2 + par)*H + tid] = wr[4*PITCH + 4 + tid];
  } else {                                      // export last interior row
    int c = tid - 256;
    slots[((wg*2 + 1)*2 + par)*H + c] = wr[67*PITCH + 4 + c];
  }
  __threadfence();
  if (PROBE && tid == 0) {
    float s = 0.0f;
    for (int w = 0; w < 16; ++w) s += scr[w];   // fixed-order sum
    energy[wg*PROBE_STEPS + (g - MELT_STEPS - 1)] = s;
  }
  __syncthreads();

  if (tid == 0) {                               // signal + wait neighbors
    __hip_atomic_store(&flags[wg], (unsigned int)g,
                       __ATOMIC_RELEASE, __HIP_MEMORY_SCOPE_AGENT);
    if (hasUp)
      while (__hip_atomic_load(&flags[wg-1], __ATOMIC_ACQUIRE,
                               __HIP_MEMORY_SCOPE_AGENT) < (unsigned int)g)
        __builtin_amdgcn_s_sleep(1);
    if (hasDn)
      while (__hip_atomic_load(&flags[wg+1], __ATOMIC_ACQUIRE,
                               __HIP_MEMORY_SCOPE_AGENT) < (unsigned int)g)
        __builtin_amdgcn_s_sleep(1);
  }
  __syncthreads();
  __threadfence();                              // make neighbor slots visible

  if (tid < 256) {                              // import ghost rows of wr
    if (hasUp) wr[3*PITCH  + 4 + tid] = slots[(((wg-1)*2 + 1)*2 + par)*H + tid];
  } else {
    int c = tid - 256;
    if (hasDn) wr[68*PITCH + 4 + c]   = slots[(((wg+1)*2 + 0)*2 + par)*H + c];
  }
  __syncthreads();
}

// float4 fill of both LDS buffers' interiors (+ rd halo rows) from an image.
__device__ __forceinline__ void load_state(const float* __restrict__ src,
                                           float* rd, float* wr,
                                           int tid, int R0, int hasUp, int hasDn)
{
  for (int i = tid; i < BAND*H/4; i += THREADS) {
    int r = i >> 6, c = (i & 63) * 4;           // 64 float4 per row
    float4 v = ((const float4*)src)[((R0 + r)*H + c) >> 2];
    *(float4*)&rd[(4+r)*PITCH + 4 + c] = v;
    *(float4*)&wr[(4+r)*PITCH + 4 + c] = v;
  }
  if (hasUp) for (int c4 = tid; c4 < H/4; c4 += THREADS)
    *(float4*)&rd[3*PITCH + 4 + c4*4]  = ((const float4*)src)[((R0-1)*H >> 2) + c4];
  if (hasDn) for (int c4 = tid; c4 < H/4; c4 += THREADS)
    *(float4*)&rd[68*PITCH + 4 + c4*4] = ((const float4*)src)[((R0+BAND)*H >> 2) + c4];
}

__global__ __launch_bounds__(THREADS, 1)
void wave_persistent(const float* __restrict__ pattern,
                     const float* __restrict__ signal,
                     float* __restrict__ out,
                     float* __restrict__ ws)
{
  __shared__ float lds[2*BUFN + 16];            // ~152 KB of the 320 KB LDS

  const int tid  = threadIdx.x;
  const int wg   = blockIdx.x;                  // 0..7
  const int img  = wg >> 2;
  const int R0   = (wg & 3) * BAND;             // first global row of band
  const int lane = tid & 31;
  const int wv   = tid >> 5;                    // wave 0..15
  const int lo   = lane & 15;
  const int hi   = lane >> 4;
  const int hasUp = (wg & 3) != 0;
  const int hasDn = (wg & 3) != 3;

  float* buf0 = lds;
  float* buf1 = lds + BUFN;
  float* scr  = lds + 2*BUFN;                   // 16 per-wave energy partials

  unsigned int* flags = reinterpret_cast<unsigned int*>(ws);
  float* energy = ws + WS_ENERGY;
  float* slots  = ws + WS_SLOTS;

  // per-lane constant WMMA operands: tridiagonal block T (symmetric, so the
  // same registers serve as A- and B-layout) and the corner selectors P/Q.
  v2f Ta[4];
#pragma unroll
  for (int c = 0; c < 4; ++c) {
    Ta[c].x = tmat(lo, 4*c + 2*hi);
    Ta[c].y = tmat(lo, 4*c + 2*hi + 1);
  }
  v2f EP; EP.x = 0.0f; EP.y = (lo == 0  && hi == 1) ? 1.0f : 0.0f;
  v2f EQ; EQ.x = (lo == 15 && hi == 0) ? 1.0f : 0.0f; EQ.y = 0.0f;

  // ---- init: zero ghost ring, load pattern, prefetch signal toward L2 ------
  for (int i = tid; i < 2*BUFN + 16; i += THREADS) lds[i] = 0.0f;
  __syncthreads();

  const float* pat = pattern + img*IMG;
  const float* sig = signal  + img*IMG;

  load_state(pat, buf0, buf1, tid, R0, hasUp, hasDn);
  for (int i = tid*16; i < BAND*H; i += THREADS*16)
    __builtin_prefetch(&sig[R0*H + i], 0, 1);   // global_prefetch_b8
  __syncthreads();

  v8f ml[4];                                    // max|lap| (melt) -> c^2 (probe)
#pragma unroll
  for (int t = 0; t < 4; ++t) ml[t] = (v8f){0,0,0,0,0,0,0,0};

  const int I  = wv >> 2;                       // tile row 0..3
  const int r0 = 4 + I*16;
  const int Jb = (wv & 3) * 4;                  // first of 4 tile cols

  int rb = 0;

  // -------- phase 1: melt & freeze (specialized loop) -----------------------
  for (int g = 1; g <= MELT_STEPS; ++g) {
    float* rd = rb ? buf1 : buf0;
    float* wr = rb ? buf0 : buf1;
    do_step<false>(g, wg, tid, lane, wv, lo, hi, hasUp, hasDn, r0, Jb,
                   rd, wr, scr, Ta, EP, EQ, ml, flags, energy, slots);
    rb ^= 1;
  }

  // -------- transition: scars = h(max|lap|), reload state with signal -------
#pragma unroll
  for (int t = 0; t < 4; ++t)
#pragma unroll
    for (int r = 0; r < 8; ++r) {
      float x     = ml[t][r];
      float inv   = 1.0f / (1.0f + 15.0f * x);
      float gamma = inv * inv;
      float scar  = (gamma < 0.4f) ? (1.0f - gamma) : 0.0f;
      ml[t][r] = 0.24f * (1.0f - scar);         // c^2 = 0.24*(1 - scars)
    }
  {
    float* rd = rb ? buf1 : buf0;
    float* wr = rb ? buf0 : buf1;
    load_state(sig, rd, wr, tid, R0, hasUp, hasDn);
    __syncthreads();
  }

  // -------- phase 2: probe (specialized loop, records energies) -------------
  for (int g = MELT_STEPS + 1; g <= TOTAL_STEPS; ++g) {
    float* rd = rb ? buf1 : buf0;
    float* wr = rb ? buf0 : buf1;
    do_step<true>(g, wg, tid, lane, wv, lo, hi, hasUp, hasDn, r0, Jb,
                  rd, wr, scr, Ta, EP, EQ, ml, flags, energy, slots);
    rb ^= 1;
  }

  // u_final = u(700) lives in bufs[rb]
  float* fin = rb ? buf1 : buf0;
  for (int i = tid; i < BAND*H/4; i += THREADS) {
    int r = i >> 6, c = (i & 63) * 4;
    ((float4*)out)[(img*IMG + (R0 + r)*H + c) >> 2] =
        *(float4*)&fin[(4+r)*PITCH + 4 + c];
  }
}

__global__ void energy_finalize(const float* __restrict__ ws,
                                float* __restrict__ out)
{
  int s = blockIdx.x * blockDim.x + threadIdx.x;
  if (s < PROBE_STEPS) {
    const float* e = ws + WS_ENERGY;
    float acc = 0.0f;
    for (int w = 0; w < WGS; ++w) acc += e[w*PROBE_STEPS + s];   // fixed order
    out[2*IMG + s] = acc;
  }
}

extern "C" void kernel_launch(void* const* d_in, const int* in_sizes, int n_in,
                              void* d_out, int out_size, void* d_ws, size_t ws_size,
                              hipStream_t stream)
{
  (void)in_sizes; (void)n_in; (void)out_size; (void)ws_size;
  const float* pattern = (const float*)d_in[0];
  const float* signal  = (const float*)d_in[1];
  // d_in[2] (3x3 laplacian) is the fixed cross stencil, baked into T/EP/EQ.
  float* out = (float*)d_out;
  float* ws  = (float*)d_ws;

  hipMemsetAsync(d_ws, 0, WGS * sizeof(unsigned int), stream);   // reset flags
  wave_persistent<<<dim3(WGS), dim3(THREADS), 0, stream>>>(pattern, signal, out, ws);
  energy_finalize<<<dim3(2), dim3(256), 0, stream>>>(ws, out);
}